// Weighted_GCN_4518305595468
// MI455X (gfx1250) — compile-verified
//
#include <hip/hip_runtime.h>
#include <hip/hip_bf16.h>

typedef float v2f __attribute__((ext_vector_type(2)));
typedef float v8f __attribute__((ext_vector_type(8)));

#define BATCH 512
#define NATOM 64
#define NAF   128
#define F1    300
#define F1P   304
#define F2    600
#define F2P   608
#define D1    512
#define D2    256
#define EPSV  1e-5f

__device__ __forceinline__ float wave_max(float v) {
#pragma unroll
  for (int o = 16; o > 0; o >>= 1) v = fmaxf(v, __shfl_xor(v, o, 32));
  return v;
}
__device__ __forceinline__ float wave_sum(float v) {
#pragma unroll
  for (int o = 16; o > 0; o >>= 1) v += __shfl_xor(v, o, 32);
  return v;
}

// mask_blank[b,n] = max_m adjs[b,n,m]
__global__ void k_mask_blank(const float* __restrict__ adjs, float* __restrict__ mb) {
  int i = blockIdx.x * blockDim.x + threadIdx.x;  // b*64+n
  if (i >= BATCH * NATOM) return;
  const float* row = adjs + (size_t)i * NATOM;
  float m = 0.f;
  for (int j = 0; j < NATOM; j++) m = fmaxf(m, row[j]);
  mb[i] = m;
}

__global__ void k_zero(float* __restrict__ p, int n) {
  int i = blockIdx.x * blockDim.x + threadIdx.x;
  if (i < n) p[i] = 0.f;
}

// One wave per (b,n) row: stream feat once, compute BOTH layer1 and layer2
// attention rows (masked softmax over 64 neighbors). dbl=1 -> Ring double softmax.
__global__ void k_attention(const float* __restrict__ feat, int C,
                            const float* __restrict__ w1, const float* __restrict__ w2,
                            const float* __restrict__ adjs, const float* __restrict__ mb,
                            float* __restrict__ A1, float* __restrict__ A2, int dbl) {
  int lane = threadIdx.x & 31;
  int wave = threadIdx.x >> 5;
  int g = blockIdx.x * 8 + wave;  // row id in [0, B*N)
  int b = g >> 6, n = g & 63;
  int m0 = lane, m1 = lane + 32;
  const float* fb = feat + ((size_t)b * C * NATOM + n) * NATOM;
  float d10 = 0.f, d11 = 0.f, d20 = 0.f, d21 = 0.f;
  for (int c = 0; c < C; c++) {
    float f0 = fb[(size_t)c * NATOM * NATOM + m0];
    float f1 = fb[(size_t)c * NATOM * NATOM + m1];
    float a = w1[c], w = w2[c];
    d10 += f0 * a; d11 += f1 * a;
    d20 += f0 * w; d21 += f1 * w;
  }
  const float* arow = adjs + (size_t)g * NATOM;
  float k0 = (1.f - arow[m0]) * -1e9f;
  float k1 = (1.f - arow[m1]) * -1e9f;
  float m2 = mb[g];

  auto smax = [&](float s0, float s1, float& p0, float& p1) {
    float mx = wave_max(fmaxf(s0, s1));
    float e0 = expf(s0 - mx), e1 = expf(s1 - mx);
    float r = 1.f / wave_sum(e0 + e1);
    p0 = e0 * r * m2;
    p1 = e1 * r * m2;
  };

  float p0, p1;
  smax(d20 + k0, d21 + k1, p0, p1);
  A2[(size_t)g * NATOM + m0] = p0;
  A2[(size_t)g * NATOM + m1] = p1;

  smax(d10 + k0, d11 + k1, p0, p1);
  if (dbl) smax(p0 + k0, p1 + k1, p0, p1);  // Ring: softmax applied twice in layer 1
  A1[(size_t)g * NATOM + m0] = p0;
  A1[(size_t)g * NATOM + m1] = p1;
}

// Fused graph-conv branch: per block (tileF, molecule b):
//  phase1: T[64,16] = X[b] @ W[:,tile]   (WMMA f32 16x16x4, 4 waves = 4 row-tiles)
//  phase2: out_tile = Adj[b] @ T          (WMMA, K=64)
//  epilogue: +bias, BN, ReLU, *mask_blank, *ave_w[i]
//  mode 0: out = e ; mode 1: out += e ; mode 2: atomic column-sum into molfp.
// Edge columns use CLAMPED-index loads (no exec divergence in the hot loop);
// pad-column garbage only affects pad-column outputs, which are never stored.
__global__ void k_branch(const float* __restrict__ Adj,
                         const float* __restrict__ X, int ldx, int K,
                         const float* __restrict__ W,
                         const float* __restrict__ bias,
                         const float* __restrict__ bnp, int F,
                         const float* __restrict__ mb,
                         const float* __restrict__ avew, int ai,
                         float* __restrict__ out, int ldo, int mode) {
  __shared__ float T[NATOM * 16];
  int lane = threadIdx.x & 31;
  int wave = threadIdx.x >> 5;  // 0..3
  int h = lane >> 4;            // half-wave: selects K-pair
  int l16 = lane & 15;
  int b = blockIdx.y;
  int col = blockIdx.x * 16 + l16;
  bool cok = col < F;
  int colc = cok ? col : (F - 1);  // clamped: valid address, garbage discarded later
  int rowA = wave * 16 + l16;

  // phase 1: T = X[b] @ W[:,tile]
  v8f acc = {};
  const float* Xa = X + (size_t)b * NATOM * ldx + (size_t)rowA * ldx + 2 * h;
  const float* Wp = W + (size_t)(2 * h) * F + colc;
  for (int kk0 = 0; kk0 < K; kk0 += 4) {
    v2f a = *(const v2f*)(Xa + kk0);   // A frag: K = v + 2h (contiguous float2)
    v2f bf;
    bf.x = Wp[0];                      // B frag: rows kk, kk+1 at fixed col
    bf.y = Wp[F];
    acc = __builtin_amdgcn_wmma_f32_16x16x4_f32(false, a, false, bf, (short)0, acc, false, false);
    Wp += 4 * (size_t)F;
  }
#pragma unroll
  for (int v = 0; v < 8; v++)
    T[(wave * 16 + v + 8 * h) * 16 + l16] = acc[v];  // D row = v + 8h
  __syncthreads();

  // phase 2: out_tile = Adj[b, rows, :] @ T
  v8f acc2 = {};
  const float* Aa = Adj + (size_t)b * NATOM * NATOM + (size_t)rowA * NATOM + 2 * h;
  const float* Tp = &T[2 * h * 16 + l16];
#pragma unroll 4
  for (int kk0 = 0; kk0 < NATOM; kk0 += 4) {
    v2f a = *(const v2f*)(Aa + kk0);
    v2f bf;
    bf.x = Tp[kk0 * 16];
    bf.y = Tp[kk0 * 16 + 16];
    acc2 = __builtin_amdgcn_wmma_f32_16x16x4_f32(false, a, false, bf, (short)0, acc2, false, false);
  }

  // epilogue (clamped parameter loads; stores remain guarded)
  float bi = bias[colc];
  float sc = bnp[colc] * rsqrtf(bnp[3 * F + colc] + EPSV);
  float mu = bnp[2 * F + colc];
  float be = bnp[F + colc];
  float aw = avew[ai];
  float csum = 0.f;
#pragma unroll
  for (int v = 0; v < 8; v++) {
    int row = wave * 16 + v + 8 * h;
    float e = acc2[v] + bi;
    e = (e - mu) * sc + be;
    e = fmaxf(e, 0.f) * mb[(size_t)b * NATOM + row] * aw;
    if (mode == 2) {
      csum += e;
    } else if (cok) {
      size_t oi = ((size_t)b * NATOM + row) * ldo + col;
      if (mode == 0) out[oi] = e; else out[oi] += e;
    }
  }
  if (mode == 2 && cok) atomicAdd(&out[(size_t)b * ldo + col], csum);
}

// eval-mode BN over channel dim (for MolFP)
__global__ void k_bn_vec(const float* __restrict__ x, const float* __restrict__ p,
                         int C, int n, float* __restrict__ y) {
  int i = blockIdx.x * blockDim.x + threadIdx.x;
  if (i >= n) return;
  int c = i % C;
  y[i] = (x[i] - p[2 * C + c]) * p[c] * rsqrtf(p[3 * C + c] + EPSV) + p[C + c];
}

// dense GEMM, 1 wave -> 16x16 tile, epilogue bias + BN + ReLU (M,N multiples of 16)
__global__ void k_gemm_dense(const float* __restrict__ A, int lda,
                             const float* __restrict__ Bm, int ldb, int K,
                             const float* __restrict__ bias,
                             const float* __restrict__ bnp, int Nn,
                             float* __restrict__ C_, int ldc) {
  int lane = threadIdx.x & 31;
  int h = lane >> 4, l16 = lane & 15;
  int col = blockIdx.x * 16 + l16;
  int rbase = blockIdx.y * 16;
  v8f acc = {};
  const float* Ap = A + (size_t)(rbase + l16) * lda + 2 * h;
  const float* Bp = Bm + (size_t)(2 * h) * ldb + col;
  for (int kk0 = 0; kk0 < K; kk0 += 4) {
    v2f a = *(const v2f*)(Ap + kk0);
    v2f bf;
    bf.x = Bp[0];
    bf.y = Bp[ldb];
    acc = __builtin_amdgcn_wmma_f32_16x16x4_f32(false, a, false, bf, (short)0, acc, false, false);
    Bp += 4 * (size_t)ldb;
  }
  float bi = bias[col];
  float sc = bnp[col] * rsqrtf(bnp[3 * Nn + col] + EPSV);
  float mu = bnp[2 * Nn + col], be = bnp[Nn + col];
#pragma unroll
  for (int v = 0; v < 8; v++) {
    int row = rbase + v + 8 * h;
    float e = (acc[v] + bi - mu) * sc + be;
    C_[(size_t)row * ldc + col] = fmaxf(e, 0.f);
  }
}

// final tiny layer: [512,256] @ [256,2] + b
__global__ void k_final(const float* __restrict__ x, const float* __restrict__ W,
                        const float* __restrict__ bias, float* __restrict__ out) {
  int t = blockIdx.x * blockDim.x + threadIdx.x;
  if (t >= BATCH * 2) return;
  int b = t >> 1, c = t & 1;
  float s = bias[c];
  for (int k = 0; k < D2; k++) s += x[(size_t)b * D2 + k] * W[k * 2 + c];
  out[t] = s;
}

extern "C" void kernel_launch(void* const* d_in, const int* in_sizes, int n_in,
                              void* d_out, int out_size, void* d_ws, size_t ws_size,
                              hipStream_t stream) {
  const float* adjs = (const float*)d_in[0];
  const float* afms = (const float*)d_in[1];
  const float* feats[5] = {(const float*)d_in[2], (const float*)d_in[3],
                           (const float*)d_in[4], (const float*)d_in[5],
                           (const float*)d_in[6]};
  const int Cs[5] = {30, 5, 3, 3, 3};
  const float* aw1[5]; const float* aw2[5];
  for (int i = 0; i < 5; i++) { aw1[i] = (const float*)d_in[7 + i]; aw2[i] = (const float*)d_in[12 + i]; }
  const float* gc1_W = (const float*)d_in[17];
  const float* gc1_b = (const float*)d_in[18];
  const float* gc2_W = (const float*)d_in[19];
  const float* gc2_b = (const float*)d_in[20];
  const float* ave1_w = (const float*)d_in[21];
  const float* ave2_w = (const float*)d_in[22];
  const float* bn1_p = (const float*)d_in[23];
  const float* bn2_p = (const float*)d_in[24];
  const float* molfp_bn = (const float*)d_in[25];
  const float* den1_W = (const float*)d_in[26];
  const float* den1_b = (const float*)d_in[27];
  const float* den2_W = (const float*)d_in[28];
  const float* den2_b = (const float*)d_in[29];
  const float* den3_W = (const float*)d_in[30];
  const float* den3_b = (const float*)d_in[31];
  const float* bnd1_p = (const float*)d_in[32];
  const float* bnd2_p = (const float*)d_in[33];

  // workspace layout (floats), ~94.3 MB total
  float* ws = (float*)d_ws;
  size_t o = 0;
  float* mb = ws + o;      o += (size_t)BATCH * NATOM;
  float* A1 = ws + o;      o += (size_t)BATCH * NATOM * NATOM;
  float* A2 = ws + o;      o += (size_t)5 * BATCH * NATOM * NATOM;
  float* x1 = ws + o;      o += (size_t)BATCH * NATOM * F1P;
  float* molfp = ws + o;   o += (size_t)BATCH * F2;
  float* molfpbn = ws + o; o += (size_t)BATCH * F2;
  float* db1 = ws + o;     o += (size_t)BATCH * D1;
  float* db2 = ws + o;     o += (size_t)BATCH * D2;

  k_mask_blank<<<BATCH * NATOM / 256, 256, 0, stream>>>(adjs, mb);
  k_zero<<<(BATCH * F2 + 255) / 256, 256, 0, stream>>>(molfp, BATCH * F2);

  // layer 1 (attention kernel also produces layer-2 attention A2[i] in the same feat pass)
  for (int i = 0; i < 5; i++) {
    k_attention<<<BATCH * NATOM / 8, 256, 0, stream>>>(
        feats[i], Cs[i], aw1[i], aw2[i], adjs, mb, A1,
        A2 + (size_t)i * BATCH * NATOM * NATOM, i == 4 ? 1 : 0);
    k_branch<<<dim3(F1P / 16, BATCH), 128, 0, stream>>>(
        A1, afms, NAF, NAF,
        gc1_W + (size_t)i * NAF * F1, gc1_b + (size_t)i * F1,
        bn1_p + (size_t)i * 4 * F1, F1, mb, ave1_w, i,
        x1, F1P, i == 0 ? 0 : 1);
  }
  // layer 2, fused with MolFP atom-sum
  for (int i = 0; i < 5; i++) {
    k_branch<<<dim3(F2P / 16, BATCH), 128, 0, stream>>>(
        A2 + (size_t)i * BATCH * NATOM * NATOM, x1, F1P, F1,
        gc2_W + (size_t)i * F1 * F2, gc2_b + (size_t)i * F2,
        bn2_p + (size_t)i * 4 * F2, F2, mb, ave2_w, i,
        molfp, F2, 2);
  }

  k_bn_vec<<<(BATCH * F2 + 255) / 256, 256, 0, stream>>>(molfp, molfp_bn, F2, BATCH * F2, molfpbn);
  k_gemm_dense<<<dim3(D1 / 16, BATCH / 16), 32, 0, stream>>>(molfpbn, F2, den1_W, D1, F2,
                                                             den1_b, bnd1_p, D1, db1, D1);
  k_gemm_dense<<<dim3(D2 / 16, BATCH / 16), 32, 0, stream>>>(db1, D1, den2_W, D2, D1,
                                                             den2_b, bnd2_p, D2, db2, D2);
  k_final<<<(BATCH * 2 + 255) / 256, 256, 0, stream>>>(db2, den3_W, den3_b, (float*)d_out);
}